// NNNet_3736621547796
// MI455X (gfx1250) — compile-verified
//
#include <hip/hip_runtime.h>
#include <math.h>

// Problem constants (fixed by the reference)
#define NN  10000
#define NE  160000
#define NG  100
#define DD  32
#define ED  6
#define QQ  96
#define CC  2
#define EPSV 1e-5f

typedef __attribute__((ext_vector_type(2))) float v2f;
typedef __attribute__((ext_vector_type(8))) float v8f;

__device__ __forceinline__ float sigmoidf_(float x) { return 1.0f / (1.0f + expf(-x)); }

__device__ __forceinline__ void atomicMaxF(float* addr, float val) {
    if (val >= 0.0f) atomicMax((int*)addr, __float_as_int(val));
    else             atomicMin((unsigned int*)addr, __float_as_uint(val));
}

// ---------------------------------------------------------------- fill
__global__ void k_fill(float* p, float v, int n) {
    int i = blockIdx.x * blockDim.x + threadIdx.x;
    if (i < n) p[i] = v;
}

// ---------------------------------------------------------------- edge MLP column stats
// relu(ea @ nw.T + nb): accumulate per-column sum / sumsq over all edges.
// 256 threads, thread owns 4 of 1024 columns; 128 edges per block (NE = 1250*128).
__global__ void k_edge_stats(const float* __restrict__ ea, const float* __restrict__ nw,
                             const float* __restrict__ nb, float* __restrict__ sum,
                             float* __restrict__ ssq) {
    __shared__ float sEA[32 * ED];
    const int tid = threadIdx.x;
    const int j0  = tid * 4;
    float w[4][ED], b[4];
    #pragma unroll
    for (int c = 0; c < 4; ++c) {
        b[c] = nb[j0 + c];
        #pragma unroll
        for (int k = 0; k < ED; ++k) w[c][k] = nw[(j0 + c) * ED + k];
    }
    float s[4] = {0,0,0,0}, q[4] = {0,0,0,0};
    const int ebase = blockIdx.x * 128;
    for (int chunk = 0; chunk < 4; ++chunk) {
        const int eb = ebase + chunk * 32;
        __syncthreads();
        if (tid < 32 * ED) sEA[tid] = ea[eb * ED + tid];
        __syncthreads();
        for (int i = 0; i < 32; ++i) {
            const float a0 = sEA[i*ED+0], a1 = sEA[i*ED+1], a2 = sEA[i*ED+2];
            const float a3 = sEA[i*ED+3], a4 = sEA[i*ED+4], a5 = sEA[i*ED+5];
            #pragma unroll
            for (int c = 0; c < 4; ++c) {
                float v = b[c];
                v = fmaf(a0, w[c][0], v); v = fmaf(a1, w[c][1], v);
                v = fmaf(a2, w[c][2], v); v = fmaf(a3, w[c][3], v);
                v = fmaf(a4, w[c][4], v); v = fmaf(a5, w[c][5], v);
                v = fmaxf(v, 0.0f);
                s[c] += v; q[c] += v * v;
            }
        }
    }
    #pragma unroll
    for (int c = 0; c < 4; ++c) {
        atomicAdd(&sum[j0 + c], s[c]);
        atomicAdd(&ssq[j0 + c], q[c]);
    }
}

// ---------------------------------------------------------------- BN stats -> scale/shift
__global__ void k_bn_finalize(const float* __restrict__ sum, const float* __restrict__ ssq,
                              const float* __restrict__ g, const float* __restrict__ beta,
                              float* __restrict__ scale, float* __restrict__ shift,
                              int ncols, float invN) {
    int j = blockIdx.x * blockDim.x + threadIdx.x;
    if (j >= ncols) return;
    float mu  = sum[j] * invN;
    float var = ssq[j] * invN - mu * mu;
    float sc  = g[j] * rsqrtf(var + EPSV);
    scale[j] = sc;
    shift[j] = beta[j] - mu * sc;
}

// ---------------------------------------------------------------- edge message + scatter
// One wave per edge; lane = output channel o. Recomputes normalized W[32,32] on the fly.
__global__ void k_edge_msg(const float* __restrict__ x, const int* __restrict__ src,
                           const int* __restrict__ dst, const float* __restrict__ ea,
                           const float* __restrict__ nw, const float* __restrict__ nb,
                           const float* __restrict__ scale, const float* __restrict__ shift,
                           float* __restrict__ agg) {
    const int wid  = (blockIdx.x * blockDim.x + threadIdx.x) >> 5;
    const int lane = threadIdx.x & 31;
    if (wid >= NE) return;
    const int s = __builtin_amdgcn_readfirstlane(src[wid]);
    const int d = __builtin_amdgcn_readfirstlane(dst[wid]);
    const float a0 = ea[wid*ED+0], a1 = ea[wid*ED+1], a2 = ea[wid*ED+2];
    const float a3 = ea[wid*ED+3], a4 = ea[wid*ED+4], a5 = ea[wid*ED+5];
    float msg = 0.0f;
    #pragma unroll 8
    for (int i = 0; i < DD; ++i) {
        const int j = i * DD + lane;
        const float* wr = nw + j * ED;
        float v = nb[j];
        v = fmaf(a0, wr[0], v); v = fmaf(a1, wr[1], v); v = fmaf(a2, wr[2], v);
        v = fmaf(a3, wr[3], v); v = fmaf(a4, wr[4], v); v = fmaf(a5, wr[5], v);
        v = fmaxf(v, 0.0f);
        v = fmaf(scale[j], v, shift[j]);
        msg = fmaf(x[s * DD + i], v, msg);
    }
    atomicAdd(&agg[d * DD + lane], msg);
}

// ---------------------------------------------------------------- node update (WMMA f32)
// xout = agg + Xin @ root + bias.  M=10000 (625 tiles), N=32 (2 tiles), K=32.
__global__ void k_node_update(const float* __restrict__ xin, const float* __restrict__ root,
                              const float* __restrict__ bias, const float* __restrict__ agg,
                              float* __restrict__ xout) {
    const int wid = (blockIdx.x * blockDim.x + threadIdx.x) >> 5;
    if (wid >= 625 * 2) return;
    const int lane = threadIdx.x & 31;
    const int half = lane >> 4;     // 0: K even pair low, 1: high
    const int l16  = lane & 15;
    const int m0 = (wid >> 1) * 16;
    const int n0 = (wid & 1) * 16;
    v8f acc = {};
    #pragma unroll
    for (int k = 0; k < DD; k += 4) {
        const int ka = k + 2 * half;
        v2f a, b;
        const float* xr = xin + (m0 + l16) * DD + ka;
        a.x = xr[0]; a.y = xr[1];
        b.x = root[ka * DD + n0 + l16];
        b.y = root[(ka + 1) * DD + n0 + l16];
        acc = __builtin_amdgcn_wmma_f32_16x16x4_f32(false, a, false, b, (short)0, acc,
                                                    false, false);
    }
    const float bv = bias[n0 + l16];
    #pragma unroll
    for (int rr = 0; rr < 8; ++rr) {
        const int row = m0 + rr + 8 * half;
        const int col = n0 + l16;
        xout[row * DD + col] = acc[rr] + agg[row * DD + col] + bv;
    }
}

// ---------------------------------------------------------------- lin1 GEMM (WMMA f32)
// pre = relu([x1|x2] @ lin1_w.T + b).  M=10000 (625 tiles), N=96 (6 tiles), K=64.
// Fused per-column sum/sumsq for the following BatchNorm.
__global__ void k_lin1_gemm(const float* __restrict__ x1, const float* __restrict__ x2,
                            const float* __restrict__ w, const float* __restrict__ bvec,
                            float* __restrict__ pre, float* __restrict__ csum,
                            float* __restrict__ csq) {
    const int wid = (blockIdx.x * blockDim.x + threadIdx.x) >> 5;
    if (wid >= 625 * 6) return;
    const int lane = threadIdx.x & 31;
    const int half = lane >> 4;
    const int l16  = lane & 15;
    const int m0 = (wid / 6) * 16;
    const int n0 = (wid % 6) * 16;
    v8f acc = {};
    #pragma unroll
    for (int k = 0; k < 2 * DD; k += 4) {
        const int ka = k + 2 * half;           // even, so ka/ka+1 in same half
        v2f a, b;
        const float* xr = (ka < DD) ? (x1 + (m0 + l16) * DD + ka)
                                    : (x2 + (m0 + l16) * DD + (ka - DD));
        a.x = xr[0]; a.y = xr[1];
        const float* wr = w + (n0 + l16) * (2 * DD) + ka;   // w[q,k], B[k][q]=w[q*64+k]
        b.x = wr[0]; b.y = wr[1];
        acc = __builtin_amdgcn_wmma_f32_16x16x4_f32(false, a, false, b, (short)0, acc,
                                                    false, false);
    }
    const float bb = bvec[n0 + l16];
    float colS = 0.0f, colQ = 0.0f;
    #pragma unroll
    for (int rr = 0; rr < 8; ++rr) {
        const int row = m0 + rr + 8 * half;
        const int col = n0 + l16;
        float v = fmaxf(acc[rr] + bb, 0.0f);
        pre[row * QQ + col] = v;
        colS += v; colQ += v * v;
    }
    colS += __shfl_xor(colS, 16, 32);
    colQ += __shfl_xor(colQ, 16, 32);
    if (half == 0) {
        atomicAdd(&csum[n0 + l16], colS);
        atomicAdd(&csq[n0 + l16], colQ);
    }
}

// ---------------------------------------------------------------- BN apply (elementwise)
__global__ void k_bn_apply(const float* __restrict__ pre, const float* __restrict__ csum,
                           const float* __restrict__ csq, const float* __restrict__ g,
                           const float* __restrict__ beta, float* __restrict__ out,
                           int rows, int cols, float invN) {
    int idx = blockIdx.x * blockDim.x + threadIdx.x;
    if (idx >= rows * cols) return;
    int col = idx % cols;
    float mu  = csum[col] * invN;
    float var = csq[col] * invN - mu * mu;
    float sc  = g[col] * rsqrtf(var + EPSV);
    out[idx] = sc * (pre[idx] - mu) + beta[col];
}

// ---------------------------------------------------------------- Set2Set LSTM step
// One block per graph, 384 threads (= 4Q gates), gates staged in LDS.
__global__ void k_lstm_step(const float* __restrict__ qs, const float* __restrict__ wih,
                            const float* __restrict__ whh, const float* __restrict__ bih,
                            const float* __restrict__ bhh, float* __restrict__ h,
                            float* __restrict__ c) {
    __shared__ float gates[4 * QQ];
    const int b = blockIdx.x;
    const int j = threadIdx.x;
    float acc = bih[j] + bhh[j];
    const float* q  = qs  + b * (2 * QQ);
    const float* wi = wih + j * (2 * QQ);
    for (int k = 0; k < 2 * QQ; ++k) acc = fmaf(q[k], wi[k], acc);
    const float* hb = h + b * QQ;
    const float* wh = whh + j * QQ;
    for (int k = 0; k < QQ; ++k) acc = fmaf(hb[k], wh[k], acc);
    gates[j] = acc;
    __syncthreads();
    if (j < QQ) {
        float ig = gates[j], fg = gates[QQ + j], gg = gates[2 * QQ + j], og = gates[3 * QQ + j];
        float cn = sigmoidf_(fg) * c[b * QQ + j] + sigmoidf_(ig) * tanhf(gg);
        float hn = sigmoidf_(og) * tanhf(cn);
        c[b * QQ + j] = cn;
        h[b * QQ + j] = hn;
    }
}

// ---------------------------------------------------------------- attention logits + seg max
__global__ void k_attn_logits(const float* __restrict__ xc, const float* __restrict__ h,
                              const int* __restrict__ batch, float* __restrict__ e,
                              float* __restrict__ emax) {
    const int wid  = (blockIdx.x * blockDim.x + threadIdx.x) >> 5;
    const int lane = threadIdx.x & 31;
    if (wid >= NN) return;
    const int b = __builtin_amdgcn_readfirstlane(batch[wid]);
    const float* xr = xc + wid * QQ;
    const float* hr = h + b * QQ;
    float v = xr[lane] * hr[lane];
    v = fmaf(xr[lane + 32], hr[lane + 32], v);
    v = fmaf(xr[lane + 64], hr[lane + 64], v);
    #pragma unroll
    for (int off = 16; off; off >>= 1) v += __shfl_down(v, off, 32);
    if (lane == 0) {
        e[wid] = v;
        atomicMaxF(&emax[b], v);
    }
}

// ---------------------------------------------------------------- exp + seg sum
__global__ void k_attn_exp(const float* __restrict__ e, const int* __restrict__ batch,
                           const float* __restrict__ emax, float* __restrict__ a,
                           float* __restrict__ asum) {
    int n = blockIdx.x * blockDim.x + threadIdx.x;
    if (n >= NN) return;
    int b = batch[n];
    float em = emax[b];
    if (!isfinite(em)) em = 0.0f;
    float av = expf(e[n] - em);
    a[n] = av;
    atomicAdd(&asum[b], av);
}

// ---------------------------------------------------------------- weighted readout r
__global__ void k_attn_r(const float* __restrict__ a, const float* __restrict__ asum,
                         const float* __restrict__ xc, const int* __restrict__ batch,
                         float* __restrict__ r) {
    const int wid  = (blockIdx.x * blockDim.x + threadIdx.x) >> 5;
    const int lane = threadIdx.x & 31;
    if (wid >= NN) return;
    const int b = __builtin_amdgcn_readfirstlane(batch[wid]);
    const float coef = a[wid] / asum[b];
    #pragma unroll
    for (int t = 0; t < 3; ++t) {
        const int q = lane + t * 32;
        atomicAdd(&r[b * QQ + q], coef * xc[wid * QQ + q]);
    }
}

// ---------------------------------------------------------------- q_star = [h | r]
__global__ void k_build_qstar(const float* __restrict__ h, const float* __restrict__ r,
                              float* __restrict__ qs) {
    int idx = blockIdx.x * blockDim.x + threadIdx.x;
    if (idx >= NG * 2 * QQ) return;
    int b = idx / (2 * QQ), j = idx % (2 * QQ);
    qs[idx] = (j < QQ) ? h[b * QQ + j] : r[b * QQ + (j - QQ)];
}

// ---------------------------------------------------------------- small dense (head)
__global__ void k_dense(const float* __restrict__ in, const float* __restrict__ w,
                        const float* __restrict__ b, float* __restrict__ out,
                        int rows, int cols, int K, int relu,
                        float* __restrict__ csum, float* __restrict__ csq) {
    int idx = blockIdx.x * blockDim.x + threadIdx.x;
    if (idx >= rows * cols) return;
    int rr = idx / cols, cc = idx % cols;
    float acc = b[cc];
    const float* ir = in + rr * K;
    const float* wr = w + cc * K;
    for (int k = 0; k < K; ++k) acc = fmaf(ir[k], wr[k], acc);
    if (relu) acc = fmaxf(acc, 0.0f);
    out[idx] = acc;
    if (csum) {
        atomicAdd(&csum[cc], acc);
        atomicAdd(&csq[cc], acc * acc);
    }
}

// ================================================================ launch
static inline int cdiv(int a, int b) { return (a + b - 1) / b; }

extern "C" void kernel_launch(void* const* d_in, const int* in_sizes, int n_in,
                              void* d_out, int out_size, void* d_ws, size_t ws_size,
                              hipStream_t stream) {
    const float* x        = (const float*)d_in[0];
    const float* ea       = (const float*)d_in[1];
    const float* c1_nn_w  = (const float*)d_in[2];
    const float* c1_nn_b  = (const float*)d_in[3];
    const float* c1_nn_g  = (const float*)d_in[4];
    const float* c1_nn_bt = (const float*)d_in[5];
    const float* c1_root  = (const float*)d_in[6];
    const float* c1_bias  = (const float*)d_in[7];
    const float* c2_nn_w  = (const float*)d_in[8];
    const float* c2_nn_b  = (const float*)d_in[9];
    const float* c2_nn_g  = (const float*)d_in[10];
    const float* c2_nn_bt = (const float*)d_in[11];
    const float* c2_root  = (const float*)d_in[12];
    const float* c2_bias  = (const float*)d_in[13];
    const float* lin1_w   = (const float*)d_in[14];
    const float* lin1_b   = (const float*)d_in[15];
    const float* lin1_g   = (const float*)d_in[16];
    const float* lin1_bt  = (const float*)d_in[17];
    const float* lstm_wih = (const float*)d_in[18];
    const float* lstm_whh = (const float*)d_in[19];
    const float* lstm_bih = (const float*)d_in[20];
    const float* lstm_bhh = (const float*)d_in[21];
    const float* s2s_w    = (const float*)d_in[22];
    const float* s2s_b    = (const float*)d_in[23];
    const float* m1_w     = (const float*)d_in[24];
    const float* m1_b     = (const float*)d_in[25];
    const float* m1_g     = (const float*)d_in[26];
    const float* m1_bt    = (const float*)d_in[27];
    const float* m2_w     = (const float*)d_in[28];
    const float* m2_b     = (const float*)d_in[29];
    const float* m2_g     = (const float*)d_in[30];
    const float* m2_bt    = (const float*)d_in[31];
    const int*   esrc     = (const int*)d_in[32];
    const int*   edst     = (const int*)d_in[33];
    const int*   batch    = (const int*)d_in[34];
    float* out = (float*)d_out;

    // workspace layout (floats)
    float* ws = (float*)d_ws;
    size_t o = 0;
    float* x1    = ws + o; o += NN * DD;          // agg1 then x1
    float* x2    = ws + o; o += NN * DD;          // agg2 then x2
    float* sum_  = ws + o; o += DD * DD;          // 1024
    float* ssq_  = ws + o; o += DD * DD;
    float* scal_ = ws + o; o += DD * DD;
    float* shft_ = ws + o; o += DD * DD;
    float* pre   = ws + o; o += NN * QQ;          // lin1 pre-BN
    float* xc    = ws + o; o += NN * QQ;
    float* csum  = ws + o; o += QQ;
    float* csq   = ws + o; o += QQ;
    float* h     = ws + o; o += NG * QQ;
    float* c     = ws + o; o += NG * QQ;          // contiguous with h
    float* r     = ws + o; o += NG * QQ;
    float* qs    = ws + o; o += NG * 2 * QQ;
    float* e     = ws + o; o += NN;
    float* a     = ws + o; o += NN;
    float* emax  = ws + o; o += NG;
    float* asum  = ws + o; o += NG;
    float* xg    = ws + o; o += NG * QQ;
    float* pm1   = ws + o; o += NG * QQ;
    float* xg2   = ws + o; o += NG * QQ;
    float* pm2   = ws + o; o += NG * CC;

    const int T = 256;

    // ------- conv1 -------
    k_fill<<<cdiv(2 * DD * DD, T), T, 0, stream>>>(sum_, 0.0f, 2 * DD * DD); // sum+ssq
    k_fill<<<cdiv(NN * DD, T), T, 0, stream>>>(x1, 0.0f, NN * DD);
    k_edge_stats<<<NE / 128, T, 0, stream>>>(ea, c1_nn_w, c1_nn_b, sum_, ssq_);
    k_bn_finalize<<<cdiv(DD * DD, T), T, 0, stream>>>(sum_, ssq_, c1_nn_g, c1_nn_bt,
                                                      scal_, shft_, DD * DD, 1.0f / NE);
    k_edge_msg<<<cdiv(NE * 32, T), T, 0, stream>>>(x, esrc, edst, ea, c1_nn_w, c1_nn_b,
                                                   scal_, shft_, x1);
    k_node_update<<<cdiv(625 * 2 * 32, T), T, 0, stream>>>(x, c1_root, c1_bias, x1, x1);

    // ------- conv2 -------
    k_fill<<<cdiv(2 * DD * DD, T), T, 0, stream>>>(sum_, 0.0f, 2 * DD * DD);
    k_fill<<<cdiv(NN * DD, T), T, 0, stream>>>(x2, 0.0f, NN * DD);
    k_edge_stats<<<NE / 128, T, 0, stream>>>(ea, c2_nn_w, c2_nn_b, sum_, ssq_);
    k_bn_finalize<<<cdiv(DD * DD, T), T, 0, stream>>>(sum_, ssq_, c2_nn_g, c2_nn_bt,
                                                      scal_, shft_, DD * DD, 1.0f / NE);
    k_edge_msg<<<cdiv(NE * 32, T), T, 0, stream>>>(x1, esrc, edst, ea, c2_nn_w, c2_nn_b,
                                                   scal_, shft_, x2);
    k_node_update<<<cdiv(625 * 2 * 32, T), T, 0, stream>>>(x1, c2_root, c2_bias, x2, x2);

    // ------- lin1 + BN -------
    k_fill<<<1, T, 0, stream>>>(csum, 0.0f, 2 * QQ);  // csum+csq contiguous
    k_lin1_gemm<<<cdiv(625 * 6 * 32, T), T, 0, stream>>>(x1, x2, lin1_w, lin1_b, pre,
                                                         csum, csq);
    k_bn_apply<<<cdiv(NN * QQ, T), T, 0, stream>>>(pre, csum, csq, lin1_g, lin1_bt, xc,
                                                   NN, QQ, 1.0f / NN);

    // ------- Set2Set -------
    k_fill<<<cdiv(2 * NG * QQ, T), T, 0, stream>>>(h, 0.0f, 2 * NG * QQ);    // h + c
    k_fill<<<cdiv(NG * 2 * QQ, T), T, 0, stream>>>(qs, 0.0f, NG * 2 * QQ);
    for (int step = 0; step < 3; ++step) {
        k_lstm_step<<<NG, 4 * QQ, 0, stream>>>(qs, lstm_wih, lstm_whh, lstm_bih, lstm_bhh,
                                               h, c);
        k_fill<<<1, T, 0, stream>>>(emax, -INFINITY, NG);
        k_fill<<<1, T, 0, stream>>>(asum, 0.0f, NG);
        k_fill<<<cdiv(NG * QQ, T), T, 0, stream>>>(r, 0.0f, NG * QQ);
        k_attn_logits<<<cdiv(NN * 32, T), T, 0, stream>>>(xc, h, batch, e, emax);
        k_attn_exp<<<cdiv(NN, T), T, 0, stream>>>(e, batch, emax, a, asum);
        k_attn_r<<<cdiv(NN * 32, T), T, 0, stream>>>(a, asum, xc, batch, r);
        k_build_qstar<<<cdiv(NG * 2 * QQ, T), T, 0, stream>>>(h, r, qs);
    }

    // ------- head -------
    k_dense<<<cdiv(NG * QQ, T), T, 0, stream>>>(qs, s2s_w, s2s_b, xg, NG, QQ, 2 * QQ, 0,
                                                nullptr, nullptr);
    k_fill<<<1, T, 0, stream>>>(csum, 0.0f, 2 * QQ);
    k_dense<<<cdiv(NG * QQ, T), T, 0, stream>>>(xg, m1_w, m1_b, pm1, NG, QQ, QQ, 1,
                                                csum, csq);
    k_bn_apply<<<cdiv(NG * QQ, T), T, 0, stream>>>(pm1, csum, csq, m1_g, m1_bt, xg2,
                                                   NG, QQ, 1.0f / NG);
    k_fill<<<1, T, 0, stream>>>(csum, 0.0f, 2 * QQ);
    k_dense<<<1, T, 0, stream>>>(xg2, m2_w, m2_b, pm2, NG, CC, QQ, 1, csum, csq);
    k_bn_apply<<<1, T, 0, stream>>>(pm2, csum, csq, m2_g, m2_bt, out, NG, CC, 1.0f / NG);
}